// PointerGeneratorNetwork_4372276707340
// MI455X (gfx1250) — compile-verified
//
#include <hip/hip_runtime.h>
#include <math.h>

#define D_MODEL 512
#define BB 4
#define SS 512
#define TT 512
#define VOCAB 32000
#define MAX_OOV 100
#define VEXT (VOCAB + MAX_OOV)   // 32100

typedef __attribute__((ext_vector_type(2))) float v2f;
typedef __attribute__((ext_vector_type(8))) float v8f;

// ---------------------------------------------------------------------------
// Fold Wo@Wc into a single 512-vector; fold bo.Wc + bc + bdo + bdi into scalar.
// ---------------------------------------------------------------------------
__global__ void fold_kernel(const float* __restrict__ Wo, const float* __restrict__ Wc,
                            const float* __restrict__ bo, const float* __restrict__ bc,
                            const float* __restrict__ bdo, const float* __restrict__ bdi,
                            float* __restrict__ wfold, float* __restrict__ gbias) {
    int e = blockIdx.x * blockDim.x + threadIdx.x;
    if (e < D_MODEL) {
        const float* row = Wo + (size_t)e * D_MODEL;
        float s = 0.f;
        for (int d = 0; d < D_MODEL; ++d) s += row[d] * Wc[d];
        wfold[e] = s;
    }
    if (e == 0) {
        float s = 0.f;
        for (int d = 0; d < D_MODEL; ++d) s += bo[d] * Wc[d];
        gbias[0] = s + bc[0] + bdo[0] + bdi[0];
    }
}

// ---------------------------------------------------------------------------
// Generic fp32 WMMA GEMM:  C = scale * (A @ B) + bias
//   A: [M,K] row-major.  BT=false: B is [K,N] row-major.  BT=true: B is [N,K].
// Each wave computes a 16(M) x 64(N) strip with V_WMMA_F32_16X16X4_F32,
// reusing the A fragment across 4 N-tiles. blockDim = 128 (4 waves).
// Grid.x * 4 waves must equal (M/16)*(N/64) exactly; grid.y = batch.
// ---------------------------------------------------------------------------
template <bool BT>
__global__ void wmma_gemm(const float* __restrict__ A, const float* __restrict__ Bm,
                          const float* __restrict__ bias, float* __restrict__ C,
                          int M, int K, int N,
                          long long sA, long long sB, long long sC, float scale) {
    int batch = blockIdx.y;
    A  += (size_t)batch * sA;
    Bm += (size_t)batch * sB;
    C  += (size_t)batch * sC;

    int wave = blockIdx.x * (blockDim.x >> 5) + (threadIdx.x >> 5);
    int lane = threadIdx.x & 31;
    int nN64 = N >> 6;
    int tm = wave / nN64;           // 16-row tile
    int tn = wave % nN64;           // 64-col tile
    int khalf = lane >> 4;          // 0: K pair {0,1}, 1: K pair {2,3}
    int mrow  = tm * 16 + (lane & 15);
    int ncol0 = tn * 64 + (lane & 15);

    const float* arow = A + (size_t)mrow * K;
    v8f c0 = {}, c1 = {}, c2 = {}, c3 = {};

    for (int k0 = 0; k0 < K; k0 += 4) {
        int kk = k0 + 2 * khalf;
        v2f a; a.x = arow[kk]; a.y = arow[kk + 1];

        v2f b0, b1, b2, b3;
        if (BT) {
            // B[N,K]: element (k, n) at Bm[n*K + k]; per-lane contiguous pair.
            const float* p0 = Bm + (size_t)(ncol0      ) * K + kk;
            const float* p1 = Bm + (size_t)(ncol0 + 16) * K + kk;
            const float* p2 = Bm + (size_t)(ncol0 + 32) * K + kk;
            const float* p3 = Bm + (size_t)(ncol0 + 48) * K + kk;
            b0.x = p0[0]; b0.y = p0[1];
            b1.x = p1[0]; b1.y = p1[1];
            b2.x = p2[0]; b2.y = p2[1];
            b3.x = p3[0]; b3.y = p3[1];
        } else {
            // B[K,N]: VGPR j holds row kk+j (lanes 0-15) / kk+j+2 via khalf.
            const float* r0 = Bm + (size_t)kk * N;
            const float* r1 = Bm + (size_t)(kk + 1) * N;
            b0.x = r0[ncol0     ]; b0.y = r1[ncol0     ];
            b1.x = r0[ncol0 + 16]; b1.y = r1[ncol0 + 16];
            b2.x = r0[ncol0 + 32]; b2.y = r1[ncol0 + 32];
            b3.x = r0[ncol0 + 48]; b3.y = r1[ncol0 + 48];
        }
        c0 = __builtin_amdgcn_wmma_f32_16x16x4_f32(false, a, false, b0, (short)0, c0, false, false);
        c1 = __builtin_amdgcn_wmma_f32_16x16x4_f32(false, a, false, b1, (short)0, c1, false, false);
        c2 = __builtin_amdgcn_wmma_f32_16x16x4_f32(false, a, false, b2, (short)0, c2, false, false);
        c3 = __builtin_amdgcn_wmma_f32_16x16x4_f32(false, a, false, b3, (short)0, c3, false, false);
    }

    // D layout: VGPR r -> row tm*16 + r + 8*khalf, col = ncol0 (+ 16*nt)
    int row0 = tm * 16 + 8 * khalf;
#pragma unroll
    for (int nt = 0; nt < 4; ++nt) {
        v8f cc = (nt == 0) ? c0 : (nt == 1) ? c1 : (nt == 2) ? c2 : c3;
        int col = ncol0 + nt * 16;
        float bv = bias ? bias[col] : 0.f;
#pragma unroll
        for (int r = 0; r < 8; ++r)
            C[(size_t)(row0 + r) * N + col] = cc[r] * scale + bv;
    }
}

// ---------------------------------------------------------------------------
// Row softmax over S=512 (one block of 512 threads per (b,t) row), in place.
// ---------------------------------------------------------------------------
__global__ void softmax_kernel(float* __restrict__ score) {
    __shared__ float red[512];
    int row = blockIdx.x, tid = threadIdx.x;
    float* p = score + (size_t)row * SS;
    float x = p[tid];
    red[tid] = x; __syncthreads();
    for (int o = 256; o > 0; o >>= 1) { if (tid < o) red[tid] = fmaxf(red[tid], red[tid + o]); __syncthreads(); }
    float mx = red[0]; __syncthreads();
    float e = __expf(x - mx);
    red[tid] = e; __syncthreads();
    for (int o = 256; o > 0; o >>= 1) { if (tid < o) red[tid] += red[tid + o]; __syncthreads(); }
    float r = e / red[0];
    p[tid] = isfinite(r) ? r : 0.f;   // nan_to_num
}

// ---------------------------------------------------------------------------
// p_gen gate: sigmoid(ctx.wfold + dec.Wdo + decin.Wdi + gbias), one row/block.
// ---------------------------------------------------------------------------
__global__ void gate_kernel(const float* __restrict__ ctx, const float* __restrict__ dec,
                            const float* __restrict__ decin, const float* __restrict__ wfold,
                            const float* __restrict__ Wdo, const float* __restrict__ Wdi,
                            const float* __restrict__ gbias, float* __restrict__ gate) {
    __shared__ float red[512];
    int row = blockIdx.x, tid = threadIdx.x;
    size_t off = (size_t)row * D_MODEL + tid;
    red[tid] = ctx[off] * wfold[tid] + dec[off] * Wdo[tid] + decin[off] * Wdi[tid];
    __syncthreads();
    for (int o = 256; o > 0; o >>= 1) { if (tid < o) red[tid] += red[tid + o]; __syncthreads(); }
    if (tid == 0) gate[row] = 1.f / (1.f + __expf(-(red[0] + gbias[0])));
}

// ---------------------------------------------------------------------------
// extend = p_gen * [vocab_dist | zeros]: float4-vectorized (VOCAB % 4 == 0).
// This is the bandwidth-dominant stage (~0.5 GB) -> b128 loads/stores.
// ---------------------------------------------------------------------------
__global__ void extend_kernel(const float* __restrict__ vocab, const float* __restrict__ gate,
                              float* __restrict__ out) {
    long long idx = (long long)blockIdx.x * blockDim.x + threadIdx.x;   // over B*T*(VEXT/4)
    int row = (int)(idx / (VEXT / 4));
    int c4  = (int)(idx % (VEXT / 4));
    float g = gate[row];
    float4 o;
    if (c4 < VOCAB / 4) {
        float4 vv = ((const float4*)vocab)[(size_t)row * (VOCAB / 4) + c4];
        o = make_float4(g * vv.x, g * vv.y, g * vv.z, g * vv.w);
    } else {
        o = make_float4(0.f, 0.f, 0.f, 0.f);
    }
    ((float4*)out)[idx] = o;
}

// ---------------------------------------------------------------------------
// Scatter-add (1 - p_gen) * score into extended dist. Duplicate src indices
// within a row require atomics -> global_atomic_add_f32.
// ---------------------------------------------------------------------------
__global__ void scatter_kernel(const int* __restrict__ src_ext, const float* __restrict__ score,
                               const float* __restrict__ gate, float* __restrict__ out) {
    long long tid = (long long)blockIdx.x * blockDim.x + threadIdx.x;   // over B*T*S
    int s = (int)(tid % SS);
    long long bt = tid / SS;
    int b = (int)(bt / TT);
    int idx = src_ext[b * SS + s];
    float val = (1.f - gate[bt]) * score[(size_t)bt * SS + s];
    atomicAdd(out + (size_t)bt * VEXT + idx, val);
}

// ---------------------------------------------------------------------------
extern "C" void kernel_launch(void* const* d_in, const int* in_sizes, int n_in,
                              void* d_out, int out_size, void* d_ws, size_t ws_size,
                              hipStream_t stream) {
    const float* enc   = (const float*)d_in[0];
    const float* dec   = (const float*)d_in[1];
    const float* vocab = (const float*)d_in[2];
    const float* decin = (const float*)d_in[3];
    const int*   srcx  = (const int*)  d_in[4];
    // d_in[5] = max_oov_len (compile-time MAX_OOV)
    const float* Wq = (const float*)d_in[6];  const float* bq  = (const float*)d_in[7];
    const float* Wk = (const float*)d_in[8];  const float* bk  = (const float*)d_in[9];
    const float* Wv = (const float*)d_in[10]; const float* bv  = (const float*)d_in[11];
    const float* Wo = (const float*)d_in[12]; const float* bo  = (const float*)d_in[13];
    const float* Wc = (const float*)d_in[14]; const float* bc  = (const float*)d_in[15];
    const float* Wdo = (const float*)d_in[16]; const float* bdo = (const float*)d_in[17];
    const float* Wdi = (const float*)d_in[18]; const float* bdi = (const float*)d_in[19];
    float* out = (float*)d_out;

    // workspace carve-up (~21 MB, all float)
    const size_t NBT = (size_t)BB * TT;           // 2048 rows
    float* ws    = (float*)d_ws;
    float* q     = ws;                            // [2048, 512]
    float* k     = q     + NBT * D_MODEL;         // [2048, 512]
    float* v     = k     + NBT * D_MODEL;         // [2048, 512]
    float* score = v     + NBT * D_MODEL;         // [B,T,S] = [2048, 512]
    float* ctx   = score + NBT * SS;              // [2048, 512]
    float* wfold = ctx   + NBT * D_MODEL;         // [512]
    float* gate  = wfold + D_MODEL;               // [2048]
    float* gbias = gate  + NBT;                   // [1]

    const float kscale = 0.04419417382415922f;    // 1/sqrt(512)

    fold_kernel<<<dim3(2), 256, 0, stream>>>(Wo, Wc, bo, bc, bdo, bdi, wfold, gbias);

    // Projections: M = B*T (or B*S) = 2048, K = N = 512. waves = 128*8 = 1024 -> 256 blocks.
    wmma_gemm<false><<<dim3(256, 1), 128, 0, stream>>>(dec, Wq, bq, q, 2048, 512, 512, 0, 0, 0, 1.f);
    wmma_gemm<false><<<dim3(256, 1), 128, 0, stream>>>(enc, Wk, bk, k, 2048, 512, 512, 0, 0, 0, 1.f);
    wmma_gemm<false><<<dim3(256, 1), 128, 0, stream>>>(enc, Wv, bv, v, 2048, 512, 512, 0, 0, 0, 1.f);

    // Attention logits: per batch q[T,D] @ k[S,D]^T, pre-scaled. waves = 32*8 = 256 -> 64 blocks.
    wmma_gemm<true><<<dim3(64, BB), 128, 0, stream>>>(q, k, nullptr, score, 512, 512, 512,
        (long long)TT * D_MODEL, (long long)SS * D_MODEL, (long long)TT * SS, kscale);

    softmax_kernel<<<dim3(BB * TT), 512, 0, stream>>>(score);

    // ctx = score @ v per batch
    wmma_gemm<false><<<dim3(64, BB), 128, 0, stream>>>(score, v, nullptr, ctx, 512, 512, 512,
        (long long)TT * SS, (long long)SS * D_MODEL, (long long)TT * D_MODEL, 1.f);

    gate_kernel<<<dim3(BB * TT), 512, 0, stream>>>(ctx, dec, decin, wfold, Wdo, Wdi, gbias, gate);

    // B*T*(VEXT/4) = 16,435,200 = 64200 * 256 exactly
    extend_kernel<<<dim3(64200), 256, 0, stream>>>(vocab, gate, out);

    // B*T*S = 1,048,576 = 4096 * 256
    scatter_kernel<<<dim3(4096), 256, 0, stream>>>(srcx, score, gate, out);
}